// CenterHeadLoss_25812753449888
// MI455X (gfx1250) — compile-verified
//
#include <hip/hip_runtime.h>

typedef __attribute__((ext_vector_type(2))) float v2f;
typedef __attribute__((ext_vector_type(4))) float v4f;
typedef __attribute__((ext_vector_type(8))) float v8f;

#define NB 512
#define NC 7
#define NH 128
#define NW 128
#define HW (NH * NW)

// ws layout (floats): [0]=pos_loss  [1]=neg_loss  [3]=reg_abs_sum  [4]=mask_sum (== num_pos)

__global__ __launch_bounds__(64) void zero_kernel(float* __restrict__ ws) {
  if (threadIdx.x < 8) ws[threadIdx.x] = 0.0f;
}

// One thread per gt box: gather 6 reg predictions at the (single) masked cell,
// accumulate L1 sum and mask count. 512 threads total -> atomics are free.
__global__ __launch_bounds__(256) void box_kernel(const float* __restrict__ preds,
                                                  const float* __restrict__ gt,
                                                  float* __restrict__ ws) {
  int b = blockIdx.x * blockDim.x + threadIdx.x;
  if (b >= NB) return;
  float cxf = gt[b * 6 + 1], cyf = gt[b * 6 + 2];
  float wv  = gt[b * 6 + 3], lv  = gt[b * 6 + 4], yaw = gt[b * 6 + 5];
  int cx = (int)floorf(cxf), cy = (int)floorf(cyf);
  if (cx < 0 || cx >= NW || cy < 0 || cy >= NH) return;   // invalid: target & mask both 0
  float tgt[6];
  tgt[0] = cxf - (float)cx;
  tgt[1] = cyf - (float)cy;
  tgt[2] = logf(wv);
  tgt[3] = logf(lv);
  tgt[4] = sinf(yaw);
  tgt[5] = cosf(yaw);
  const float* pr = preds + (size_t)b * NC * HW + HW + cy * NW + cx;  // channels 1..6
  float s = 0.0f;
#pragma unroll
  for (int c = 0; c < 6; ++c) s += fabsf(pr[c * HW] - tgt[c]);
  atomicAdd(&ws[3], s);
  atomicAdd(&ws[4], 1.0f);
}

constexpr int TPB = 256;          // 8 waves
constexpr int EPT = 8;            // elements per thread (two b128 loads)
constexpr int EPB = TPB * EPT;    // 2048 elements per block
constexpr int BPB = HW / EPB;     // 8 blocks per batch image

// Streaming focal-loss reduction over the heatmap channel.
__global__ __launch_bounds__(TPB) void hm_kernel(const float* __restrict__ preds,
                                                 const float* __restrict__ gt,
                                                 float* __restrict__ ws) {
  const int blk   = blockIdx.x;
  const int b     = blk >> 3;               // / BPB
  const int chunk = (blk & (BPB - 1)) * EPB;
  const int tid   = threadIdx.x;

  // Per-batch box center (uniform across the block; L2-resident broadcast load).
  float cxf = gt[b * 6 + 1], cyf = gt[b * 6 + 2];
  int cx = (int)floorf(cxf), cy = (int)floorf(cyf);
  bool valid = (cx >= 0) & (cx < NW) & (cy >= 0) & (cy < NH);

  // Read-once data: non-temporal 128-bit loads, 1 KB contiguous per wave.
  const float* src = preds + (size_t)b * NC * HW + chunk + tid * EPT;
  v4f p0 = __builtin_nontemporal_load((const v4f*)src);
  v4f p1 = __builtin_nontemporal_load((const v4f*)src + 1);

  float z[8] = {p0.x, p0.y, p0.z, p0.w, p1.x, p1.y, p1.z, p1.w};
  float negl = 0.0f, posl = 0.0f;
  const int base = chunk + tid * EPT;
#pragma unroll
  for (int e = 0; e < 8; ++e) {
    int idx = base + e;
    int y = idx >> 7;
    int x = idx & (NW - 1);
    float p = 1.0f / (1.0f + __expf(-z[e]));
    // Analytic target heatmap: 1.0 at center, 0.8 at in-range 8-neighbors, else 0.
    float t = 0.0f;
    int dy = y - cy, dx = x - cx;
    if (valid && dy >= -1 && dy <= 1 && dx >= -1 && dx <= 1)
      t = ((dy | dx) == 0) ? 1.0f : 0.8f;
    if (t == 1.0f) {
      float omp = 1.0f - p;
      posl += __logf(p) * omp * omp;
    } else {
      float omt = 1.0f - t;          // 1.0 or 0.2
      float nw  = omt * omt; nw *= nw;
      negl += __logf(1.0f - p) * (p * p) * nw;
    }
  }

  // --- wave32 reduction on the matrix pipe -------------------------------
  // D = A(16x4) * ones(4x16) + 0 : D[m,n] = rowsum(m), where A's row m pulls
  // slots {lane m V0, lane m V1, lane m+16 V0, lane m+16 V1}. With partial in
  // V0 and 0 in V1: lane 0 holds rows 0..7 in its 8 D-VGPRs (lanes 0..7 +
  // 16..23), lane 16 holds rows 8..15; one shfl_xor(16) completes the sum.
  // B = all-ones, so B's exact slot layout is irrelevant.
  v2f a0;  a0.x = negl; a0.y = 0.0f;
  v2f a1;  a1.x = posl; a1.y = 0.0f;
  v2f one; one.x = 1.0f; one.y = 1.0f;
  v8f acc = {};
  v8f dn = __builtin_amdgcn_wmma_f32_16x16x4_f32(false, a0, false, one, (short)0, acc, false, false);
  v8f dp = __builtin_amdgcn_wmma_f32_16x16x4_f32(false, a1, false, one, (short)0, acc, false, false);
  float sn = dn[0] + dn[1] + dn[2] + dn[3] + dn[4] + dn[5] + dn[6] + dn[7];
  float sp = dp[0] + dp[1] + dp[2] + dp[3] + dp[4] + dp[5] + dp[6] + dp[7];
  sn += __shfl_xor(sn, 16, 32);
  sp += __shfl_xor(sp, 16, 32);

  __shared__ float red[16];
  int wave = tid >> 5, lane = tid & 31;
  if (lane == 0) { red[wave] = sn; red[8 + wave] = sp; }
  __syncthreads();
  if (tid == 0) {
    float tn = 0.0f, tp = 0.0f;
#pragma unroll
    for (int w = 0; w < 8; ++w) { tn += red[w]; tp += red[8 + w]; }
    atomicAdd(&ws[1], tn);
    atomicAdd(&ws[0], tp);
  }
}

__global__ void fin_kernel(const float* __restrict__ ws, float* __restrict__ out) {
  float pos  = ws[0], neg = ws[1], reg = ws[3], msum = ws[4];
  float npos = msum;  // #cells with target==1.0 == #valid boxes == sum(mask)
  float loss_hm  = (npos > 0.0f) ? (-(pos + neg) / fmaxf(npos, 1.0f)) : (-neg);
  float loss_reg = reg / (msum + 0.0001f);
  out[0] = loss_hm + 2.0f * loss_reg;
}

extern "C" void kernel_launch(void* const* d_in, const int* in_sizes, int n_in,
                              void* d_out, int out_size, void* d_ws, size_t ws_size,
                              hipStream_t stream) {
  const float* preds = (const float*)d_in[0];
  const float* gt    = (const float*)d_in[1];
  float* ws = (float*)d_ws;
  zero_kernel<<<1, 64, 0, stream>>>(ws);
  box_kernel<<<(NB + 255) / 256, 256, 0, stream>>>(preds, gt, ws);
  hm_kernel<<<NB * BPB, TPB, 0, stream>>>(preds, gt, ws);
  fin_kernel<<<1, 1, 0, stream>>>(ws, (float*)d_out);
}